// SAGEConv_5214090297415
// MI455X (gfx1250) — compile-verified
//
#include <hip/hip_runtime.h>

typedef __attribute__((ext_vector_type(2))) float f32x2;
typedef __attribute__((ext_vector_type(4))) float f32x4;
typedef __attribute__((ext_vector_type(8))) float f32x8;

#define D 256
#define LDS_STRIDE 260   // 256 + 4 pad floats: rows hit distinct LDS banks

// ---------------------------------------------------------------------------
// Kernel 1: zero the scratch accumulators (summed[N*256] ++ deg[N]).
// ---------------------------------------------------------------------------
__global__ void zero_ws_kernel(float* __restrict__ ws, long total) {
    long base = ((long)blockIdx.x * blockDim.x + threadIdx.x) * 4;
#pragma unroll
    for (int j = 0; j < 4; ++j) {
        long i = base + j;
        if (i < total) ws[i] = 0.0f;
    }
}

// ---------------------------------------------------------------------------
// Kernel 2: edge scatter.  64 threads per edge; thread c handles float4 chunk
// c of the 256-wide feature row.  Coalesced float4 gather of feat[src], four
// global_atomic_add_f32 into summed[dst]; chunk 0 also counts the degree.
// ---------------------------------------------------------------------------
__global__ void scatter_kernel(const float* __restrict__ feat,
                               const int*   __restrict__ src,
                               const int*   __restrict__ dst,
                               float*       __restrict__ summed,
                               float*       __restrict__ deg,
                               long n_edges) {
    long gid = (long)blockIdx.x * blockDim.x + threadIdx.x;
    long e   = gid >> 6;
    if (e >= n_edges) return;
    int c = (int)(gid & 63);

    int s = src[e];
    int d = dst[e];

    f32x4 v = *(const f32x4*)(feat + (long)s * D + c * 4);
    float* o = summed + (long)d * D + c * 4;
    atomicAdd(o + 0, v.x);
    atomicAdd(o + 1, v.y);
    atomicAdd(o + 2, v.z);
    atomicAdd(o + 3, v.w);
    if (c == 0) atomicAdd(deg + d, 1.0f);
}

// ---------------------------------------------------------------------------
// Kernel 3: fused dual GEMM via V_WMMA_F32_16X16X4_F32 (exact fp32 path).
//   out = feat @ w_self^T + b_self + (summed/max(deg,1)) @ w_neigh^T
// One block (8 waves) per 16-row M-tile.  Mean division fused into the LDS
// staging store.  Each wave owns two 16-col N-tiles; K=256 in steps of 4.
// ---------------------------------------------------------------------------
__launch_bounds__(256, 2)
__global__ void sage_gemm_kernel(const float* __restrict__ feat,
                                 const float* __restrict__ summed,
                                 const float* __restrict__ deg,
                                 const float* __restrict__ w_self,
                                 const float* __restrict__ w_neigh,
                                 const float* __restrict__ b_self,
                                 float*       __restrict__ out,
                                 int n_nodes) {
    __shared__ float ldsA[16 * LDS_STRIDE];  // feat tile
    __shared__ float ldsS[16 * LDS_STRIDE];  // mean-scaled neighbor tile

    const int m0  = blockIdx.x * 16;
    const int tid = threadIdx.x;

    // ---- Phase 1: cooperative, fully coalesced float4 staging into LDS ----
#pragma unroll
    for (int i = 0; i < 4; ++i) {
        int q   = tid + i * 256;      // 1024 float4 slots per tile
        int row = q >> 6;             // 0..15
        int c4  = q & 63;             // float4 column
        int m   = m0 + row;
        int mc  = (m < n_nodes) ? m : (n_nodes - 1);  // clamp (tail tile safety)

        f32x4 f = *(const f32x4*)(feat   + (long)mc * D + c4 * 4);
        f32x4 s = *(const f32x4*)(summed + (long)mc * D + c4 * 4);
        float dg    = deg[mc];
        float scale = 1.0f / fmaxf(dg, 1.0f);   // mean; isolated nodes -> 0/1
        s *= scale;

        *(f32x4*)(ldsA + row * LDS_STRIDE + c4 * 4) = f;
        *(f32x4*)(ldsS + row * LDS_STRIDE + c4 * 4) = s;
    }
    __syncthreads();

    // ---- Phase 2: WMMA.  fp32 16x16x4 operand layout:
    //   A: lane = M row (0..15); lane-half selects K pair {0,1}/{2,3};
    //      VGPR0/1 = the two consecutive K values  -> one ds_load_b64.
    //   B: lane = N col; same K-pair split          -> one global b64 load.
    const int lane = tid & 31;
    const int wave = tid >> 5;
    const int half = lane >> 4;        // 0 -> K%4 in {0,1}, 1 -> {2,3}
    const int r16  = lane & 15;

    const int ldsRow = r16 * LDS_STRIDE + half * 2;

#pragma unroll
    for (int t = 0; t < 2; ++t) {
        const int nt = wave * 2 + t;           // N-tile 0..15
        const int n  = nt * 16 + r16;          // output column for this lane

        const float* wS = w_self  + (long)n * D + half * 2;
        const float* wN = w_neigh + (long)n * D + half * 2;

        // Seed accumulator with the bias (same column n for all 8 acc rows).
        f32x8 acc;
        {
            float bias = b_self[n];
#pragma unroll
            for (int r = 0; r < 8; ++r) acc[r] = bias;
        }

        // feat @ w_self^T
#pragma unroll 8
        for (int k0 = 0; k0 < D; k0 += 4) {
            f32x2 a = *(const f32x2*)(ldsA + ldsRow + k0);
            f32x2 b = *(const f32x2*)(wS + k0);
            acc = __builtin_amdgcn_wmma_f32_16x16x4_f32(
                false, a, false, b, (short)0, acc, false, false);
        }
        // mean(neigh) @ w_neigh^T
#pragma unroll 8
        for (int k0 = 0; k0 < D; k0 += 4) {
            f32x2 a = *(const f32x2*)(ldsS + ldsRow + k0);
            f32x2 b = *(const f32x2*)(wN + k0);
            acc = __builtin_amdgcn_wmma_f32_16x16x4_f32(
                false, a, false, b, (short)0, acc, false, false);
        }

        // D layout: VGPR r, lane half h -> M = m0 + 8*h + r, N = n.
#pragma unroll
        for (int r = 0; r < 8; ++r) {
            int m = m0 + half * 8 + r;
            if (m < n_nodes) out[(long)m * D + n] = acc[r];
        }
    }
}

// ---------------------------------------------------------------------------
// Host launcher.  Inputs: 0=feat f32, 1=src i32, 2=dst i32, 3=w_neigh f32,
// 4=w_self f32, 5=b_self f32.  d_ws holds summed[N*256] ++ deg[N] (f32).
// ---------------------------------------------------------------------------
extern "C" void kernel_launch(void* const* d_in, const int* in_sizes, int n_in,
                              void* d_out, int out_size, void* d_ws, size_t ws_size,
                              hipStream_t stream) {
    const float* feat    = (const float*)d_in[0];
    const int*   src     = (const int*)  d_in[1];
    const int*   dst     = (const int*)  d_in[2];
    const float* w_neigh = (const float*)d_in[3];
    const float* w_self  = (const float*)d_in[4];
    const float* b_self  = (const float*)d_in[5];
    float*       out     = (float*)d_out;

    const long n_nodes = in_sizes[0] / D;
    const long n_edges = in_sizes[1];

    float* summed = (float*)d_ws;
    float* deg    = summed + n_nodes * D;

    // 1) zero accumulators
    {
        long total = n_nodes * (D + 1);
        long blk   = (total / 4 + 255) / 256;
        zero_ws_kernel<<<(unsigned)blk, 256, 0, stream>>>(summed, total);
    }
    // 2) edge scatter (64 threads / edge)
    {
        long total = n_edges * 64;
        long blk   = (total + 255) / 256;
        scatter_kernel<<<(unsigned)blk, 256, 0, stream>>>(feat, src, dst,
                                                          summed, deg, n_edges);
    }
    // 3) fused dual WMMA GEMM + bias + mean
    {
        unsigned mtiles = (unsigned)((n_nodes + 15) / 16);
        sage_gemm_kernel<<<mtiles, 256, 0, stream>>>(feat, summed, deg,
                                                     w_self, w_neigh, b_self,
                                                     out, (int)n_nodes);
    }
}